// gvp_emb_80539226735229
// MI455X (gfx1250) — compile-verified
//
#include <hip/hip_runtime.h>

#define EPS_C    1e-8f
#define LN_EPS_C 1e-5f

// node-state strides (padded for aligned vector access)
#define SST 8    // s_cur stride (floats); rows 32B-aligned
#define VST 12   // v_cur stride (floats); rows 48B -> 16B-aligned

typedef float v2f __attribute__((ext_vector_type(2)));
typedef float f4  __attribute__((ext_vector_type(4)));
typedef float v8f __attribute__((ext_vector_type(8)));

// butterfly XOR across 32 lanes via ds_swizzle (group-of-32: and=0x1f, or=0, xor=m)
#define SWZ_XOR(x, m) __int_as_float(__builtin_amdgcn_ds_swizzle(__float_as_int(x), (((m) << 10) | 0x1f)))

__device__ __forceinline__ float sigmoidf_(float x) { return 1.f / (1.f + __expf(-x)); }

// LayerNorm-tuple on (s[6], v[3,3]) in-place
__device__ __forceinline__ void ln_tuple_63(const float* __restrict__ g, const float* __restrict__ b,
                                            float s[6], float v[9]) {
  float mu = 0.f;
#pragma unroll
  for (int k = 0; k < 6; ++k) mu += s[k];
  mu *= (1.f / 6.f);
  float var = 0.f;
#pragma unroll
  for (int k = 0; k < 6; ++k) { float d = s[k] - mu; var += d * d; }
  var *= (1.f / 6.f);
  float inv = rsqrtf(var + LN_EPS_C);
#pragma unroll
  for (int k = 0; k < 6; ++k) s[k] = (s[k] - mu) * inv * g[k] + b[k];
  float m = 0.f;
#pragma unroll
  for (int i = 0; i < 3; ++i) {
    float n2 = v[i*3]*v[i*3] + v[i*3+1]*v[i*3+1] + v[i*3+2]*v[i*3+2];
    m += fmaxf(n2, EPS_C);
  }
  float r = rsqrtf(m * (1.f / 3.f));
#pragma unroll
  for (int k = 0; k < 9; ++k) v[k] *= r;
}

// GVP with si=6, vi=3, so=6, vo=3 (h=3). wh[3,3], ws_w[6,9], ws_b[6], wv[3,3]
__device__ __forceinline__ void gvp_633(const float* __restrict__ wh, const float* __restrict__ ws_w,
                                        const float* __restrict__ ws_b, const float* __restrict__ wv,
                                        const float s_in[6], const float v_in[9],
                                        float s_out[6], float v_out[9],
                                        bool relu_act, bool sig_gate) {
  float vh[9];  // vh[j*3+d] = sum_i v_in[i*3+d] * wh[j*3+i]
#pragma unroll
  for (int j = 0; j < 3; ++j)
#pragma unroll
    for (int d = 0; d < 3; ++d) {
      float a = 0.f;
#pragma unroll
      for (int i = 0; i < 3; ++i) a += v_in[i*3+d] * wh[j*3+i];
      vh[j*3+d] = a;
    }
  float vn[3];
#pragma unroll
  for (int j = 0; j < 3; ++j)
    vn[j] = sqrtf(fmaxf(vh[j*3]*vh[j*3] + vh[j*3+1]*vh[j*3+1] + vh[j*3+2]*vh[j*3+2], EPS_C));
#pragma unroll
  for (int o = 0; o < 6; ++o) {
    float a = ws_b[o];
#pragma unroll
    for (int k = 0; k < 6; ++k) a += s_in[k] * ws_w[o*9+k];
#pragma unroll
    for (int j = 0; j < 3; ++j) a += vn[j] * ws_w[o*9+6+j];
    s_out[o] = relu_act ? fmaxf(a, 0.f) : a;
  }
#pragma unroll
  for (int jo = 0; jo < 3; ++jo) {
    float x = 0.f, y = 0.f, z = 0.f;
#pragma unroll
    for (int i = 0; i < 3; ++i) {
      float w = wv[jo*3+i];
      x += vh[i*3+0] * w; y += vh[i*3+1] * w; z += vh[i*3+2] * w;
    }
    if (sig_gate) {
      float nr = sqrtf(fmaxf(x*x + y*y + z*z, EPS_C));
      float sg = sigmoidf_(nr);
      x *= sg; y *= sg; z *= sg;
    }
    v_out[jo*3+0] = x; v_out[jo*3+1] = y; v_out[jo*3+2] = z;
  }
}

// ---- node state load/store helpers (vectorized; padded strides) ----
__device__ __forceinline__ void load_s(const float* __restrict__ s_cur, long n, float s[6]) {
  const float* p = s_cur + n * SST;
  f4  a = *(const f4*)(p);
  v2f b = *(const v2f*)(p + 4);
  s[0] = a[0]; s[1] = a[1]; s[2] = a[2]; s[3] = a[3]; s[4] = b[0]; s[5] = b[1];
}
__device__ __forceinline__ void store_s(float* __restrict__ s_cur, long n, const float s[6]) {
  float* p = s_cur + n * SST;
  f4 a = {s[0], s[1], s[2], s[3]};
  v2f b = {s[4], s[5]};
  *(f4*)(p) = a; *(v2f*)(p + 4) = b;
}
__device__ __forceinline__ void load_v(const float* __restrict__ v_cur, long n, float v[9]) {
  const float* p = v_cur + n * VST;
  f4 a = *(const f4*)(p);
  f4 b = *(const f4*)(p + 4);
  v[0]=a[0]; v[1]=a[1]; v[2]=a[2]; v[3]=a[3];
  v[4]=b[0]; v[5]=b[1]; v[6]=b[2]; v[7]=b[3];
  v[8]=p[8];
}
__device__ __forceinline__ void store_v(float* __restrict__ v_cur, long n, const float v[9]) {
  float* p = v_cur + n * VST;
  f4 a = {v[0], v[1], v[2], v[3]};
  f4 b = {v[4], v[5], v[6], v[7]};
  *(f4*)(p) = a; *(f4*)(p + 4) = b; p[8] = v[8];
}

__global__ __launch_bounds__(256) void zero_f(float* __restrict__ p, long n) {
  long i = (long)blockIdx.x * blockDim.x + threadIdx.x;
  if (i < n) p[i] = 0.f;
}

__global__ void scan_offsets(const int* __restrict__ blen, int* __restrict__ off, int G) {
  if (threadIdx.x == 0 && blockIdx.x == 0) {
    int a = 0;
    for (int g = 0; g < G; ++g) { off[g] = a; a += blen[g]; }
  }
}

__global__ __launch_bounds__(256) void count_deg(const int* __restrict__ eidx, float* __restrict__ cnt, int E) {
  int e = blockIdx.x * blockDim.x + threadIdx.x;
  if (e < E) atomicAdd(&cnt[eidx[E + e]], 1.0f);
}

// ln_v + W_v  (per node)
__global__ __launch_bounds__(256) void node_embed(
    const float* __restrict__ hVs, const float* __restrict__ hVv,
    const float* __restrict__ lnV_g, const float* __restrict__ lnV_b,
    const float* __restrict__ wh, const float* __restrict__ ws_w,
    const float* __restrict__ ws_b, const float* __restrict__ wv,
    float* __restrict__ s_cur, float* __restrict__ v_cur, int N) {
  int n = blockIdx.x * blockDim.x + threadIdx.x;
  if (n >= N) return;
  float s[6], v[9];
  {  // hVs rows: 6 floats, 8B-aligned
    const float* p = hVs + (long)n * 6;
    v2f a = *(const v2f*)p, b = *(const v2f*)(p + 2), c = *(const v2f*)(p + 4);
    s[0]=a[0]; s[1]=a[1]; s[2]=b[0]; s[3]=b[1]; s[4]=c[0]; s[5]=c[1];
  }
#pragma unroll
  for (int k = 0; k < 9; ++k) v[k] = hVv[(long)n*9+k];
  ln_tuple_63(lnV_g, lnV_b, s, v);
  float so[6], vo[9];
  gvp_633(wh, ws_w, ws_b, wv, s, v, so, vo, false, false);
  store_s(s_cur, n, so);
  store_v(v_cur, n, vo);
}

// Per-edge: fused (ln_e + W_e) edge embedding, then m0->m1->m2 message MLP, atomic scatter to dst
__global__ __launch_bounds__(256) void conv_kernel(
    const int* __restrict__ eidx,
    const float* __restrict__ hEs, const float* __restrict__ hEv,
    const float* __restrict__ s_cur, const float* __restrict__ v_cur,
    float* __restrict__ agg_s, float* __restrict__ agg_v,
    const float* __restrict__ lnE_g, const float* __restrict__ lnE_b,
    const float* __restrict__ We_wh, const float* __restrict__ We_w,
    const float* __restrict__ We_b, const float* __restrict__ We_wv,
    const float* __restrict__ m0_wh, const float* __restrict__ m0_w,
    const float* __restrict__ m0_b, const float* __restrict__ m0_wv,
    const float* __restrict__ m1_wh, const float* __restrict__ m1_w,
    const float* __restrict__ m1_b, const float* __restrict__ m1_wv,
    const float* __restrict__ m2_wh, const float* __restrict__ m2_w,
    const float* __restrict__ m2_b, const float* __restrict__ m2_wv,
    int E) {
  int e = blockIdx.x * blockDim.x + threadIdx.x;
  if (e >= E) return;
  int src = eidx[e];
  int dst = eidx[E + e];

  // ---- edge embedding: ln_e on (es[32], ev[1,3]) + GVP W_e (32,1 -> 32,1) ----
  float es[32];
  {  // 128B-aligned rows -> 8 x b128 loads
    const f4* ep = (const f4*)(hEs + (long)e * 32);
#pragma unroll
    for (int i = 0; i < 8; ++i) {
      f4 q = ep[i];
      es[i*4+0] = q[0]; es[i*4+1] = q[1]; es[i*4+2] = q[2]; es[i*4+3] = q[3];
    }
  }
  float ev0 = hEv[(long)e*3 + 0], ev1 = hEv[(long)e*3 + 1], ev2 = hEv[(long)e*3 + 2];
  float mu = 0.f;
#pragma unroll
  for (int k = 0; k < 32; ++k) mu += es[k];
  mu *= (1.f / 32.f);
  float var = 0.f;
#pragma unroll
  for (int k = 0; k < 32; ++k) { float d = es[k] - mu; var += d * d; }
  var *= (1.f / 32.f);
  float inv = rsqrtf(var + LN_EPS_C);
#pragma unroll
  for (int k = 0; k < 32; ++k) es[k] = (es[k] - mu) * inv * lnE_g[k] + lnE_b[k];
  float vr = rsqrtf(fmaxf(ev0*ev0 + ev1*ev1 + ev2*ev2, EPS_C));  // vi=1: mean == value
  ev0 *= vr; ev1 *= vr; ev2 *= vr;
  float whE = We_wh[0];
  float vhE0 = ev0 * whE, vhE1 = ev1 * whE, vhE2 = ev2 * whE;
  float vnE = sqrtf(fmaxf(vhE0*vhE0 + vhE1*vhE1 + vhE2*vhE2, EPS_C));
  float esc[32];
#pragma unroll
  for (int o = 0; o < 32; ++o) {
    float a = We_b[o];
#pragma unroll
    for (int k = 0; k < 32; ++k) a += es[k] * We_w[o*33 + k];
    a += vnE * We_w[o*33 + 32];
    esc[o] = a;
  }
  float wvE = We_wv[0];
  float evc0 = wvE * vhE0, evc1 = wvE * vhE1, evc2 = wvE * vhE2;

  // ---- gather endpoint node states (vectorized, L2-resident) ----
  float ss[6], sd[6], vs[9], vd[9];
  load_s(s_cur, src, ss); load_s(s_cur, dst, sd);
  load_v(v_cur, src, vs); load_v(v_cur, dst, vd);

  // ---- m0: GVP(si=44, vi=7 -> so=6, vo=3), relu + sigmoid gate ----
  float mv[21];  // rows: vs(3), evc(1), vd(3); mv[i*3+d]
#pragma unroll
  for (int k = 0; k < 9; ++k) mv[k] = vs[k];
  mv[9] = evc0; mv[10] = evc1; mv[11] = evc2;
#pragma unroll
  for (int k = 0; k < 9; ++k) mv[12 + k] = vd[k];
  float vh0[21];  // vh0[j*3+d] = sum_{i<7} mv[i*3+d]*wh[j*7+i]
#pragma unroll
  for (int j = 0; j < 7; ++j)
#pragma unroll
    for (int d = 0; d < 3; ++d) {
      float a = 0.f;
#pragma unroll
      for (int i = 0; i < 7; ++i) a += mv[i*3+d] * m0_wh[j*7+i];
      vh0[j*3+d] = a;
    }
  float vn0[7];
#pragma unroll
  for (int j = 0; j < 7; ++j)
    vn0[j] = sqrtf(fmaxf(vh0[j*3]*vh0[j*3] + vh0[j*3+1]*vh0[j*3+1] + vh0[j*3+2]*vh0[j*3+2], EPS_C));
  float s1[6];
#pragma unroll
  for (int o = 0; o < 6; ++o) {
    float a = m0_b[o];
#pragma unroll
    for (int k = 0; k < 6; ++k)  a += ss[k]  * m0_w[o*51 + k];
#pragma unroll
    for (int k = 0; k < 32; ++k) a += esc[k] * m0_w[o*51 + 6 + k];
#pragma unroll
    for (int k = 0; k < 6; ++k)  a += sd[k]  * m0_w[o*51 + 38 + k];
#pragma unroll
    for (int j = 0; j < 7; ++j)  a += vn0[j] * m0_w[o*51 + 44 + j];
    s1[o] = fmaxf(a, 0.f);
  }
  float v1[9];
#pragma unroll
  for (int jo = 0; jo < 3; ++jo) {
    float x = 0.f, y = 0.f, z = 0.f;
#pragma unroll
    for (int i = 0; i < 7; ++i) {
      float w = m0_wv[jo*7 + i];
      x += vh0[i*3+0] * w; y += vh0[i*3+1] * w; z += vh0[i*3+2] * w;
    }
    float nr = sqrtf(fmaxf(x*x + y*y + z*z, EPS_C));
    float sg = sigmoidf_(nr);
    v1[jo*3+0] = x * sg; v1[jo*3+1] = y * sg; v1[jo*3+2] = z * sg;
  }

  // ---- m1 (relu+gate), m2 (no act) ----
  float s2[6], v2[9], s3[6], v3[9];
  gvp_633(m1_wh, m1_w, m1_b, m1_wv, s1, v1, s2, v2, true, true);
  gvp_633(m2_wh, m2_w, m2_b, m2_wv, s2, v2, s3, v3, false, false);

  // ---- scatter-add (mean denominator applied in node_update) ----
#pragma unroll
  for (int o = 0; o < 6; ++o) atomicAdd(&agg_s[(long)dst*6 + o], s3[o]);
#pragma unroll
  for (int k = 0; k < 9; ++k) atomicAdd(&agg_v[(long)dst*9 + k], v3[k]);
}

// residual + ln0 + ff0 + ff1 + residual + ln1 (per node, in place)
__global__ __launch_bounds__(256) void node_update(
    float* __restrict__ s_cur, float* __restrict__ v_cur,
    const float* __restrict__ agg_s, const float* __restrict__ agg_v, const float* __restrict__ cnt,
    const float* __restrict__ ln0_g, const float* __restrict__ ln0_b,
    const float* __restrict__ ff0_wh, const float* __restrict__ ff0_w,
    const float* __restrict__ ff0_b,  const float* __restrict__ ff0_wv,
    const float* __restrict__ ff1_wh, const float* __restrict__ ff1_w,
    const float* __restrict__ ff1_b,  const float* __restrict__ ff1_wv,
    const float* __restrict__ ln1_g, const float* __restrict__ ln1_b, int N) {
  int n = blockIdx.x * blockDim.x + threadIdx.x;
  if (n >= N) return;
  float rc = 1.f / fmaxf(cnt[n], 1.f);
  float s[6], v[9];
  load_s(s_cur, n, s);
  load_v(v_cur, n, v);
#pragma unroll
  for (int k = 0; k < 6; ++k) s[k] += agg_s[(long)n*6+k] * rc;
#pragma unroll
  for (int k = 0; k < 9; ++k) v[k] += agg_v[(long)n*9+k] * rc;
  ln_tuple_63(ln0_g, ln0_b, s, v);

  // ff0: GVP(6,3 -> 24,6), h=6, relu + gate
  float vh[18];
#pragma unroll
  for (int j = 0; j < 6; ++j)
#pragma unroll
    for (int d = 0; d < 3; ++d) {
      float a = 0.f;
#pragma unroll
      for (int i = 0; i < 3; ++i) a += v[i*3+d] * ff0_wh[j*3+i];
      vh[j*3+d] = a;
    }
  float vn[6];
#pragma unroll
  for (int j = 0; j < 6; ++j)
    vn[j] = sqrtf(fmaxf(vh[j*3]*vh[j*3] + vh[j*3+1]*vh[j*3+1] + vh[j*3+2]*vh[j*3+2], EPS_C));
  float fs[24];
#pragma unroll
  for (int o = 0; o < 24; ++o) {
    float a = ff0_b[o];
#pragma unroll
    for (int k = 0; k < 6; ++k) a += s[k]  * ff0_w[o*12 + k];
#pragma unroll
    for (int j = 0; j < 6; ++j) a += vn[j] * ff0_w[o*12 + 6 + j];
    fs[o] = fmaxf(a, 0.f);
  }
  float fv[18];
#pragma unroll
  for (int jo = 0; jo < 6; ++jo) {
    float x = 0.f, y = 0.f, z = 0.f;
#pragma unroll
    for (int i = 0; i < 6; ++i) {
      float w = ff0_wv[jo*6 + i];
      x += vh[i*3+0] * w; y += vh[i*3+1] * w; z += vh[i*3+2] * w;
    }
    float nr = sqrtf(fmaxf(x*x + y*y + z*z, EPS_C));
    float sg = sigmoidf_(nr);
    fv[jo*3+0] = x * sg; fv[jo*3+1] = y * sg; fv[jo*3+2] = z * sg;
  }
  // ff1: GVP(24,6 -> 6,3), h=6, no act
  float vh2[18];
#pragma unroll
  for (int j = 0; j < 6; ++j)
#pragma unroll
    for (int d = 0; d < 3; ++d) {
      float a = 0.f;
#pragma unroll
      for (int i = 0; i < 6; ++i) a += fv[i*3+d] * ff1_wh[j*6+i];
      vh2[j*3+d] = a;
    }
  float vn2[6];
#pragma unroll
  for (int j = 0; j < 6; ++j)
    vn2[j] = sqrtf(fmaxf(vh2[j*3]*vh2[j*3] + vh2[j*3+1]*vh2[j*3+1] + vh2[j*3+2]*vh2[j*3+2], EPS_C));
#pragma unroll
  for (int o = 0; o < 6; ++o) {
    float a = ff1_b[o];
#pragma unroll
    for (int k = 0; k < 24; ++k) a += fs[k]  * ff1_w[o*30 + k];
#pragma unroll
    for (int j = 0; j < 6; ++j)  a += vn2[j] * ff1_w[o*30 + 24 + j];
    s[o] += a;  // residual on post-ln0 s
  }
#pragma unroll
  for (int jo = 0; jo < 3; ++jo) {
    float x = 0.f, y = 0.f, z = 0.f;
#pragma unroll
    for (int i = 0; i < 6; ++i) {
      float w = ff1_wv[jo*6 + i];
      x += vh2[i*3+0] * w; y += vh2[i*3+1] * w; z += vh2[i*3+2] * w;
    }
    v[jo*3+0] += x; v[jo*3+1] += y; v[jo*3+2] += z;
  }
  ln_tuple_63(ln1_g, ln1_b, s, v);
  store_s(s_cur, n, s);
  store_v(v_cur, n, v);
}

// ln_out + W_out (6,3 -> 6,0 relu), scatter into padded [G, max_len, 6] buffer
__global__ __launch_bounds__(256) void node_out(
    const float* __restrict__ s_cur, const float* __restrict__ v_cur,
    const float* __restrict__ lnO_g, const float* __restrict__ lnO_b,
    const float* __restrict__ wh, const float* __restrict__ ws_w, const float* __restrict__ ws_b,
    const int* __restrict__ batch, const int* __restrict__ offsets, int max_len,
    float* __restrict__ out_pad, int N) {
  int n = blockIdx.x * blockDim.x + threadIdx.x;
  if (n >= N) return;
  float s[6], v[9];
  load_s(s_cur, n, s);
  load_v(v_cur, n, v);
  ln_tuple_63(lnO_g, lnO_b, s, v);
  float vh[9];
#pragma unroll
  for (int j = 0; j < 3; ++j)
#pragma unroll
    for (int d = 0; d < 3; ++d) {
      float a = 0.f;
#pragma unroll
      for (int i = 0; i < 3; ++i) a += v[i*3+d] * wh[j*3+i];
      vh[j*3+d] = a;
    }
  float vn[3];
#pragma unroll
  for (int j = 0; j < 3; ++j)
    vn[j] = sqrtf(fmaxf(vh[j*3]*vh[j*3] + vh[j*3+1]*vh[j*3+1] + vh[j*3+2]*vh[j*3+2], EPS_C));
  int g = batch[n];
  int pos = n - offsets[g];
  if (pos < 0 || pos >= max_len) return;
  long row = (long)g * max_len + pos;
  float o6[6];
#pragma unroll
  for (int o = 0; o < 6; ++o) {
    float a = ws_b[o];
#pragma unroll
    for (int k = 0; k < 6; ++k) a += s[k]  * ws_w[o*9 + k];
#pragma unroll
    for (int j = 0; j < 3; ++j) a += vn[j] * ws_w[o*9 + 6 + j];
    o6[o] = fmaxf(a, 0.f);
  }
  float* p = out_pad + row * 6;  // 8B-aligned
  v2f w0 = {o6[0], o6[1]}, w1 = {o6[2], o6[3]}, w2 = {o6[4], o6[5]};
  *(v2f*)(p) = w0; *(v2f*)(p + 2) = w1; *(v2f*)(p + 4) = w2;
}

// Dense Linear(6,64) + LayerNorm(64) over all padded rows via V_WMMA_F32_16X16X4_F32.
// One wave = one 16-row M-tile; 4 N-tiles of 16; K=6 padded to 8 -> 2 k-steps.
__global__ __launch_bounds__(128) void dense_ln_wmma(
    const float* __restrict__ out_pad,  // [R,6]
    const float* __restrict__ dw,       // [64,6]
    const float* __restrict__ db,       // [64]
    const float* __restrict__ lg, const float* __restrict__ lb,  // LN gamma/beta [64]
    float* __restrict__ h, int R) {
  int wave = (blockIdx.x * blockDim.x + threadIdx.x) >> 5;
  int lane = threadIdx.x & 31;
  long row0 = (long)wave * 16;
  if (row0 >= R) return;           // wave-uniform exit; EXEC stays full for live waves
  int mrow = lane & 15;
  bool hi = lane >= 16;
  int kbase = hi ? 2 : 0;          // K_local = vgpr + 2*(lane>=16)
  const v2f vzero = {0.f, 0.f};

  // B (weights): lane holds B[k][n], n = nt*16+mrow, k = ks*4 + kbase(+1).
  // k-step 1, high half would be k=6,7 -> structurally zero. All loads 8B-aligned.
  v2f bmat[4][2];
#pragma unroll
  for (int nt = 0; nt < 4; ++nt) {
    const float* np = dw + (nt * 16 + mrow) * 6;
    bmat[nt][0] = *(const v2f*)(np + kbase);
    v2f t = *(const v2f*)(np + 4);
    bmat[nt][1] = hi ? vzero : t;
  }
  // A (activations): lane holds A[M][k], M = mrow; same K padding scheme.
  long arow = row0 + mrow;
  bool rowok = arow < R;
  const float* ap = out_pad + (rowok ? arow : 0) * 6;
  v2f amat0 = *(const v2f*)(ap + kbase);
  v2f amat1 = *(const v2f*)(ap + 4);
  if (!rowok) amat0 = vzero;
  if (!rowok || hi) amat1 = vzero;

  v8f acc[4];
#pragma unroll
  for (int nt = 0; nt < 4; ++nt) {
    v8f c = {};
    c = __builtin_amdgcn_wmma_f32_16x16x4_f32(false, amat0, false, bmat[nt][0], (short)0, c, false, false);
    c = __builtin_amdgcn_wmma_f32_16x16x4_f32(false, amat1, false, bmat[nt][1], (short)0, c, false, false);
    acc[nt] = c;
  }
  // + bias (n = nt*16 + mrow, same for all 8 rows in the VGPR)
#pragma unroll
  for (int nt = 0; nt < 4; ++nt) {
    float bias = db[nt*16 + mrow];
#pragma unroll
    for (int r = 0; r < 8; ++r) acc[nt][r] += bias;
  }
  // LayerNorm(64) per row: row values live in 16 lanes of this half x 4 nt.
  // Butterfly reduction via ds_swizzle (xor<=8 keeps the two halves independent).
#pragma unroll
  for (int r = 0; r < 8; ++r) {
    float sm = 0.f, sq = 0.f;
#pragma unroll
    for (int nt = 0; nt < 4; ++nt) { float x = acc[nt][r]; sm += x; sq += x * x; }
    sm += SWZ_XOR(sm, 1);  sq += SWZ_XOR(sq, 1);
    sm += SWZ_XOR(sm, 2);  sq += SWZ_XOR(sq, 2);
    sm += SWZ_XOR(sm, 4);  sq += SWZ_XOR(sq, 4);
    sm += SWZ_XOR(sm, 8);  sq += SWZ_XOR(sq, 8);
    float mean = sm * (1.f / 64.f);
    float var  = sq * (1.f / 64.f) - mean * mean;
    float inv  = rsqrtf(var + LN_EPS_C);
    long row = row0 + r + (hi ? 8 : 0);
    if (row < R) {
#pragma unroll
      for (int nt = 0; nt < 4; ++nt) {
        int n = nt*16 + mrow;
        h[row*64 + n] = (acc[nt][r] - mean) * inv * lg[n] + lb[n];  // coalesced 16-lane rows
      }
    }
  }
}

__global__ void graph_sum(const float* __restrict__ h, float* __restrict__ sums, int max_len) {
  int g = blockIdx.x;
  int c = threadIdx.x;  // 64 threads
  const float* p = h + (long)g * max_len * 64 + c;
  float acc = 0.f;
  for (int r = 0; r < max_len; ++r) acc += p[(long)r * 64];
  sums[(long)g * 64 + c] = acc;
}

extern "C" void kernel_launch(void* const* d_in, const int* in_sizes, int n_in,
                              void* d_out, int out_size, void* d_ws, size_t ws_size,
                              hipStream_t stream) {
  const float* hVs  = (const float*)d_in[0];
  const float* hVv  = (const float*)d_in[1];
  const int*   eidx = (const int*)d_in[2];
  const float* hEs  = (const float*)d_in[3];
  const float* hEv  = (const float*)d_in[4];
  const int*   batch= (const int*)d_in[5];
  const int*   blen = (const int*)d_in[6];
  const int N = in_sizes[0] / 6;
  const int E = in_sizes[3] / 32;
  const int G = in_sizes[6];
  const int max_len = out_size / (G * 64) - 1;  // out = h[G,ml,64] ++ sum[G,64]
  const long R = (long)G * max_len;

#define PF(i) ((const float*)d_in[(i)])
  // pytree (sorted-dict) parameter leaf indices
  // 8..11: W_e.{wh, ws.b, ws.w, wv}; 12..14: W_out.{wh, ws.b, ws.w}
  // 15..18: W_v.{wh, ws.b, ws.w, wv}; 19/20: dense.{b,w}; 21/22: dense_ln.{b,g}
  // layer l base: 23 + 24*l -> conv.m0(4), m1(4), m2(4), ff0(4), ff1(4), ln0.{b,g}, ln1.{b,g}
  // 95/96: ln_e.{b,g}; 97/98: ln_out.{b,g}; 99/100: ln_v.{b,g}

  float* w = (float*)d_ws;
  float* s_cur   = w;  w += (size_t)N * SST;
  float* v_cur   = w;  w += (size_t)N * VST;
  float* agg_s   = w;  w += (size_t)N * 6;
  float* agg_v   = w;  w += (size_t)N * 9;
  float* cnt     = w;  w += (size_t)N;
  float* out_pad = w;  w += (size_t)R * 6;
  int*   offsets = (int*)w;

  const int TB = 256;
  auto nb = [](long x, int t) { return (unsigned)((x + t - 1) / t); };

  zero_f<<<nb(R * 6, TB), TB, 0, stream>>>(out_pad, R * 6);
  zero_f<<<nb(N, TB), TB, 0, stream>>>(cnt, N);
  scan_offsets<<<1, 32, 0, stream>>>(blen, offsets, G);
  node_embed<<<nb(N, TB), TB, 0, stream>>>(hVs, hVv, PF(100), PF(99),
                                           PF(15), PF(17), PF(16), PF(18),
                                           s_cur, v_cur, N);
  count_deg<<<nb(E, TB), TB, 0, stream>>>(eidx, cnt, E);

  for (int l = 0; l < 3; ++l) {
    int L = 23 + l * 24;
    zero_f<<<nb((long)N * 15, TB), TB, 0, stream>>>(agg_s, (long)N * 15);  // agg_s + agg_v contiguous
    conv_kernel<<<nb(E, TB), TB, 0, stream>>>(
        eidx, hEs, hEv, s_cur, v_cur, agg_s, agg_v,
        PF(96), PF(95),
        PF(8), PF(10), PF(9), PF(11),
        PF(L + 0), PF(L + 2), PF(L + 1), PF(L + 3),
        PF(L + 4), PF(L + 6), PF(L + 5), PF(L + 7),
        PF(L + 8), PF(L + 10), PF(L + 9), PF(L + 11), E);
    node_update<<<nb(N, TB), TB, 0, stream>>>(
        s_cur, v_cur, agg_s, agg_v, cnt,
        PF(L + 21), PF(L + 20),
        PF(L + 12), PF(L + 14), PF(L + 13), PF(L + 15),
        PF(L + 16), PF(L + 18), PF(L + 17), PF(L + 19),
        PF(L + 23), PF(L + 22), N);
  }

  node_out<<<nb(N, TB), TB, 0, stream>>>(s_cur, v_cur, PF(98), PF(97),
                                         PF(12), PF(14), PF(13),
                                         batch, offsets, max_len, out_pad, N);

  long tiles = (R + 15) / 16;          // one wave per 16-row tile, 4 waves/block
  unsigned blocks = (unsigned)((tiles + 3) / 4);
  dense_ln_wmma<<<blocks, 128, 0, stream>>>(out_pad, PF(20), PF(19), PF(22), PF(21),
                                            (float*)d_out, (int)R);
  graph_sum<<<G, 64, 0, stream>>>((const float*)d_out, (float*)d_out + R * 64, max_len);
#undef PF
}